// DynamicAttention_45268955300289
// MI455X (gfx1250) — compile-verified
//
#include <hip/hip_runtime.h>
#include <hip/hip_bf16.h>
#include <cstddef>

typedef float v2f __attribute__((ext_vector_type(2)));
typedef float v8f __attribute__((ext_vector_type(8)));

#define B_  8
#define K_  1024
#define L_  1024
#define D_  1024
#define O_  6
#define E_  128
#define OE_ 768   // O_*E_

// ---------------------------------------------------------------------------
// Kernel 1: priors GEMM in exact fp32 via V_WMMA_F32_16X16X4_F32.
//   priors[bk][o][e] = sum_d U[bk][d] * W[o][d][e]
// Per-o GEMM: (8192 x 1024) x (1024 x 128). Block = 256 thr (8 waves);
// each wave owns a 16x128 strip (8 accumulator tiles). Grid = (64, 6).
// Fragment layouts (CDNA5 ISA 7.12.2, 32-bit 16x4 / 4x16 / 16x16):
//   A: lane l, vgpr j -> A[l%16][2*(l/16)+j]   (contiguous -> one b64 load)
//   B: lane l, vgpr j -> B[2*(l/16)+j][l%16]
//   D: lane l, vgpr r -> D[r + 8*(l/16)][l%16]
// ---------------------------------------------------------------------------
__global__ __launch_bounds__(256) void priors_gemm(const float* __restrict__ U,
                                                   const float* __restrict__ W,
                                                   float* __restrict__ P)
{
    const int o    = blockIdx.y;
    const int wave = threadIdx.x >> 5;
    const int lane = threadIdx.x & 31;
    const int half = lane >> 4;      // which half-wave (K split)
    const int mn   = lane & 15;      // A row / B col within tile
    const int m0   = blockIdx.x * 128 + wave * 16;

    const float* __restrict__ Wo = W + (size_t)o * (D_ * E_);
    const float* __restrict__ Ar = U + (size_t)(m0 + mn) * D_ + 2 * half;

    const v8f vzero = {0.f, 0.f, 0.f, 0.f, 0.f, 0.f, 0.f, 0.f};
    v8f acc[8];
#pragma unroll
    for (int s = 0; s < 8; ++s) acc[s] = vzero;

    for (int k0 = 0; k0 < D_; k0 += 4) {
        v2f a = *(const v2f*)(Ar + k0);                      // A[mn][k0+2h .. +1]
        const float* Bp = Wo + (size_t)(k0 + 2 * half) * E_ + mn;
#pragma unroll
        for (int s = 0; s < 8; ++s) {
            v2f bf;
            bf.x = Bp[s * 16];          // row k0+2h,   col s*16+mn
            bf.y = Bp[s * 16 + E_];     // row k0+2h+1, col s*16+mn
            acc[s] = __builtin_amdgcn_wmma_f32_16x16x4_f32(
                false, a, false, bf, (short)0, acc[s], false, false);
        }
    }

    float* Pb = P + (size_t)m0 * OE_ + o * E_;
#pragma unroll
    for (int s = 0; s < 8; ++s) {
#pragma unroll
        for (int r = 0; r < 8; ++r) {
            const int m = r + half * 8;
            Pb[(size_t)m * OE_ + s * 16 + mn] = acc[s][r];
        }
    }
}

// ---------------------------------------------------------------------------
// Kernel 2: dynamic routing with the L dimension collapsed (results are
// provably identical for every l). One workgroup per b; logits/probs live in
// LDS (~52 KB of 320 KB). 1024 threads = 32 waves, thread t <-> capsule k=t.
// ---------------------------------------------------------------------------
__global__ __launch_bounds__(1024) void routing_kernel(const float* __restrict__ P,
                                                       const int*   __restrict__ mask,
                                                       float* __restrict__ probs_ws,
                                                       float* __restrict__ v_ws)
{
    const int b = blockIdx.x;
    const int t = threadIdx.x;

    __shared__ float lg[K_ * O_];    // logits[k][o]   24 KB
    __shared__ float pr[K_ * O_];    // probs[k][o]    24 KB
    __shared__ float sv[O_ * E_];    // s then v        3 KB
    __shared__ float red[24];        // per-wave partials for squash
    __shared__ float scale[O_];

    for (int i = t; i < K_ * O_; i += 1024) lg[i] = 0.f;
    const bool masked = (mask[b * K_ + t] != 0);
    const float* __restrict__ Pb = P + (size_t)b * K_ * OE_;
    __syncthreads();

    for (int it = 0; it < 3; ++it) {
        // ---- probs[k][*] = softmax_o(mask ? -1e18 : logits[k][*]) ----------
        {
            float l[O_];
#pragma unroll
            for (int oo = 0; oo < O_; ++oo) l[oo] = lg[t * O_ + oo];
            if (masked) {
#pragma unroll
                for (int oo = 0; oo < O_; ++oo) pr[t * O_ + oo] = 1.f / 6.f;
            } else {
                float mx = l[0];
#pragma unroll
                for (int oo = 1; oo < O_; ++oo) mx = fmaxf(mx, l[oo]);
                float ex[O_]; float sum = 0.f;
#pragma unroll
                for (int oo = 0; oo < O_; ++oo) { ex[oo] = expf(l[oo] - mx); sum += ex[oo]; }
#pragma unroll
                for (int oo = 0; oo < O_; ++oo) pr[t * O_ + oo] = ex[oo] / sum;
            }
        }
        __syncthreads();

        // ---- s[o][e] = sum_k probs[k][o] * priors[k][o][e] ----------------
        // 768 active threads: t -> (o,e); loads coalesced across e.
        if (t < O_ * E_) {
            const int oo = t >> 7, ee = t & 127;
            const float* pp = Pb + oo * E_ + ee;
            const float* pw = &pr[oo];
            float acc = 0.f;
            for (int k = 0; k < K_; ++k)
                acc += pw[k * O_] * pp[(size_t)k * OE_];
            sv[t] = acc;
        }
        __syncthreads();

        // ---- squash over e (128 = 4 full waves per o) ----------------------
        float p = (t < O_ * E_) ? sv[t] * sv[t] : 0.f;
#pragma unroll
        for (int off = 16; off; off >>= 1) p += __shfl_xor(p, off, 32);
        if (t < O_ * E_ && (t & 31) == 0) red[t >> 5] = p;
        __syncthreads();
        if (t < O_) {
            const float sq = red[t * 4] + red[t * 4 + 1] + red[t * 4 + 2] + red[t * 4 + 3];
            scale[t] = sq / (1.f + sq) / (sqrtf(sq) + 1e-8f);
        }
        __syncthreads();
        if (t < O_ * E_) sv[t] *= scale[t >> 7];          // sv now holds v
        __syncthreads();

        // ---- logits[k][o] += dot_e(priors[k][o][:], v[o][:]) (iters 0,1) --
        if (it < 2) {
            const int wv = t >> 5, ln = t & 31;
            for (int k = wv; k < K_; k += 32) {
                const float* pk = Pb + (size_t)k * OE_;
#pragma unroll
                for (int oo = 0; oo < O_; ++oo) {
                    float d = 0.f;
#pragma unroll
                    for (int j = 0; j < 4; ++j) {
                        const int ee = ln + 32 * j;       // coalesced across lanes
                        d += pk[oo * E_ + ee] * sv[oo * E_ + ee];
                    }
#pragma unroll
                    for (int off = 16; off; off >>= 1) d += __shfl_xor(d, off, 32);
                    if (ln == 0) lg[k * O_ + oo] += d;
                }
            }
        }
        __syncthreads();
    }

    for (int i = t; i < K_ * O_; i += 1024) probs_ws[(size_t)b * K_ * O_ + i] = pr[i];
    if (t < O_ * E_) v_ws[(size_t)b * O_ * E_ + t] = sv[t];
}

// ---------------------------------------------------------------------------
// Kernel 3: broadcast over L (the 226 MB write — the true roofline cost).
// One block per (b,l); float4 stores, sources are L2-hot (~220 KB total).
// ---------------------------------------------------------------------------
__global__ __launch_bounds__(256) void broadcast_kernel(const float* __restrict__ probs_ws,
                                                        const float* __restrict__ v_ws,
                                                        float* __restrict__ out)
{
    const int bl = blockIdx.x;           // b*L + l, 0..8191
    const int b  = bl >> 10;

    float4*       vd = (float4*)(out + (size_t)bl * OE_);
    const float4* vs = (const float4*)(v_ws + (size_t)b * OE_);
    for (int i = threadIdx.x; i < OE_ / 4; i += 256) vd[i] = vs[i];

    float4*       pd = (float4*)(out + (size_t)B_ * L_ * OE_ + (size_t)bl * (K_ * O_));
    const float4* ps = (const float4*)(probs_ws + (size_t)b * (K_ * O_));
    for (int i = threadIdx.x; i < (K_ * O_) / 4; i += 256) pd[i] = ps[i];
}

// ---------------------------------------------------------------------------
extern "C" void kernel_launch(void* const* d_in, const int* in_sizes, int n_in,
                              void* d_out, int out_size, void* d_ws, size_t ws_size,
                              hipStream_t stream)
{
    const float* U    = (const float*)d_in[0];   // inputs_u        (8,1024,1024)
    // d_in[1] (context_sequence) is never read by the reference computation.
    const float* W    = (const float*)d_in[2];   // route_weights   (6,1024,128)
    const int*   mask = (const int*)  d_in[3];   // inputs_mask     (8,1024)

    float* out = (float*)d_out;
    // Stage priors (25 MB) in the probs_c region of d_out (201 MB): it is
    // only consumed before broadcast_kernel overwrites all of d_out.
    float* priors   = out + (size_t)B_ * L_ * OE_;
    // Small scratch (~220 KB) in d_ws: broadcast must read these while it
    // rewrites every byte of d_out.
    float* probs_ws = (float*)d_ws;                       // B*K*O  = 49152 f
    float* v_ws     = probs_ws + (size_t)B_ * K_ * O_;    // B*O*E  =  6144 f

    dim3 g1(64, 6);
    priors_gemm<<<g1, 256, 0, stream>>>(U, W, priors);
    routing_kernel<<<B_, 1024, 0, stream>>>(priors, mask, probs_ws, v_ws);
    broadcast_kernel<<<B_ * L_, 256, 0, stream>>>(probs_ws, v_ws, out);
}